// TTTLayer_11570641895534
// MI455X (gfx1250) — compile-verified
//
#include <hip/hip_runtime.h>
#include <hip/hip_bf16.h>

typedef float v2f __attribute__((ext_vector_type(2)));
typedef float v4f __attribute__((ext_vector_type(4)));
typedef float v8f __attribute__((ext_vector_type(8)));
typedef int   v4i __attribute__((ext_vector_type(4)));

typedef __attribute__((address_space(1))) v4i GlobalV4i;  // global (AS1) int4
typedef __attribute__((address_space(3))) v4i SharedV4i;  // LDS (AS3) int4

#define WAVES_PER_BLOCK 8
#define THREADS_PER_BLOCK 256

#if __has_builtin(__builtin_amdgcn_global_load_async_to_lds_b128)
#define USE_ASYNC_LDS 1
#else
#define USE_ASYNC_LDS 0
#endif

// Issue the copy of 576 contiguous floats (2 positions) into a wave-private LDS slab.
// Async path: 5 GLOBAL_LOAD_ASYNC_TO_LDS_B128 issues (ASYNCcnt += 5), no VGPR round-trip.
__device__ __forceinline__ void stage_issue(const float* __restrict__ g, float* l, int lane) {
#if USE_ASYNC_LDS
    #pragma unroll
    for (int i = 0; i < 5; ++i) {
        const int idx = lane + i * 32;
        if (idx < 144) {
            __builtin_amdgcn_global_load_async_to_lds_b128(
                (GlobalV4i*)(g + idx * 4),
                (SharedV4i*)(l + idx * 4),
                /*imm offset*/0, /*cpol*/0);
        }
    }
#else
    const v4f* __restrict__ s4 = (const v4f*)g;
    v4f* d4 = (v4f*)l;
    #pragma unroll
    for (int i = 0; i < 5; ++i) {
        const int idx = lane + i * 32;
        if (idx < 144) d4[idx] = s4[idx];
    }
#endif
}

// Wait until the CURRENT buffer's staging is complete.
// pendingNext!=0 means 5 async ops for the next buffer were just issued and may remain.
__device__ __forceinline__ void stage_wait(int pendingNext) {
#if USE_ASYNC_LDS
  #if __has_builtin(__builtin_amdgcn_s_wait_asynccnt)
    if (pendingNext) __builtin_amdgcn_s_wait_asynccnt(5);
    else             __builtin_amdgcn_s_wait_asynccnt(0);
    asm volatile("" ::: "memory");
  #else
    if (pendingNext) asm volatile("s_wait_asynccnt 0x5" ::: "memory");
    else             asm volatile("s_wait_asynccnt 0x0" ::: "memory");
  #endif
#else
    (void)pendingNext;
    asm volatile("s_wait_dscnt 0x0" ::: "memory");
#endif
}

// samples: [P][9][32] fp32 (P = 131072, 288 floats/position, pairs are contiguous)
// kernelW: [96][32] fp32 row-major (shared); staged transposed (KT[c][r]) in LDS
// out:     [P][256] fp32: out[p][line*32 + f]
__global__ __launch_bounds__(THREADS_PER_BLOCK)
void ttt_lines_wmma(const float* __restrict__ samples,
                    const float* __restrict__ kernelW,
                    float* __restrict__ out,
                    int npairs)
{
    __shared__ float sKT[32 * 96];                     // 12 KB
    __shared__ float sA[WAVES_PER_BLOCK][2][576];      // 36 KB, double-buffered per wave

    const int tid  = threadIdx.x;
    const int wid  = tid >> 5;
    const int lane = tid & 31;

    for (int i = tid; i < 96 * 32; i += THREADS_PER_BLOCK) {
        sKT[(i & 31) * 96 + (i >> 5)] = kernelW[i];    // KT[c][r] = K[r][c]
    }
    __syncthreads();

    const int m_n  = lane & 15;   // A row (mod 16) / B,C column
    const int half = lane >> 4;   // K sub-pair select / output position select
    const int line = m_n & 7;

    // LINES[line][0..2] without table lookups
    int c0, c1, c2;
    if (line < 3)       { c0 = line * 3; c1 = c0 + 1; c2 = c0 + 2; }
    else if (line < 6)  { c0 = line - 3; c1 = c0 + 3; c2 = c0 + 6; }
    else if (line == 6) { c0 = 0; c1 = 4; c2 = 8; }
    else                { c0 = 2; c1 = 4; c2 = 6; }

    const int aBase = (m_n >> 3) * 288 + half * 2;     // + cell*32 + (ks&7)*4 per step
    const int bBase = m_n * 96 + half * 2;

    // Hoist loop-invariant B fragments into registers: 24 ksteps x 2 N-halves x v2f = 96 VGPRs.
    v2f breg0[24], breg1[24];
    #pragma unroll
    for (int ks = 0; ks < 24; ++ks) {
        breg0[ks] = *(const v2f*)(sKT + bBase + ks * 4);
        breg1[ks] = *(const v2f*)(sKT + bBase + 16 * 96 + ks * 4);
    }

    const int stride = gridDim.x * WAVES_PER_BLOCK;
    int pair = blockIdx.x * WAVES_PER_BLOCK + wid;
    int cur = 0;

    if (pair < npairs) stage_issue(samples + (size_t)pair * 576, &sA[wid][0][0], lane);

    for (; pair < npairs; pair += stride, cur ^= 1) {
        const int nxt = pair + stride;
        float* curBuf = &sA[wid][cur][0];
        if (nxt < npairs) {
            stage_issue(samples + (size_t)nxt * 576, &sA[wid][cur ^ 1][0], lane);
            stage_wait(1);
        } else {
            stage_wait(0);
        }

        // 16x32 output tile: (16x96 gathered A) x (96x32 kernel), 24 K-steps
        v8f acc0 = {};
        v8f acc1 = {};
        #pragma unroll
        for (int ks = 0; ks < 24; ++ks) {
            const int cell = (ks < 8) ? c0 : ((ks < 16) ? c1 : c2);
            const v2f a = *(const v2f*)(curBuf + aBase + cell * 32 + (ks & 7) * 4);
            acc0 = __builtin_amdgcn_wmma_f32_16x16x4_f32(
                false, a, false, breg0[ks], (short)0, acc0, false, false);
            acc1 = __builtin_amdgcn_wmma_f32_16x16x4_f32(
                false, a, false, breg1[ks], (short)0, acc1, false, false);
        }

        // C/D VGPR r: lanes 0-15 -> M=r (pos 2*pair), lanes 16-31 -> M=r+8 (pos 2*pair+1)
        float* dst = out + (size_t)(2 * pair + half) * 256 + m_n;
        #pragma unroll
        for (int r = 0; r < 8; ++r) {
            dst[r * 32]      = acc0[r];
            dst[r * 32 + 16] = acc1[r];
        }
    }
}

extern "C" void kernel_launch(void* const* d_in, const int* in_sizes, int n_in,
                              void* d_out, int out_size, void* d_ws, size_t ws_size,
                              hipStream_t stream) {
    const float* samples = (const float*)d_in[0];
    const float* kernelW = (const float*)d_in[1];
    float* out = (float*)d_out;

    const int positions = in_sizes[0] / (9 * 32);  // 131072
    const int npairs = positions / 2;              // 65536

    int blocks = (npairs + WAVES_PER_BLOCK - 1) / WAVES_PER_BLOCK;
    if (blocks > 2048) blocks = 2048;              // grid-stride: ~4 pairs per wave

    ttt_lines_wmma<<<blocks, THREADS_PER_BLOCK, 0, stream>>>(samples, kernelW, out, npairs);
}